// GATlayer_73598559584327
// MI455X (gfx1250) — compile-verified
//
#include <hip/hip_runtime.h>
#include <hip/hip_bf16.h>
#include <math.h>

#define SLOT  2
#define DIM   128
#define ROW   (SLOT * DIM)   /* 256 floats per node */
#define SLOPE 0.98f

typedef float v2f __attribute__((ext_vector_type(2)));
typedef float v8f __attribute__((ext_vector_type(8)));

// ---- monotone float <-> uint mapping for atomic max on floats ----
__device__ __forceinline__ unsigned flip_f32(float f) {
    unsigned b = __float_as_uint(f);
    return (b & 0x80000000u) ? ~b : (b | 0x80000000u);
}
__device__ __forceinline__ float unflip_f32(unsigned u) {
    unsigned b = (u & 0x80000000u) ? (u & 0x7fffffffu) : ~u;
    return __uint_as_float(b);
}

// ---------------- phase 0: zero output accumulator + per-node state --------
__global__ void k_init(float4* __restrict__ out4, int nOut4,
                       unsigned* __restrict__ m_u, float* __restrict__ s, int N) {
    int i      = blockIdx.x * blockDim.x + threadIdx.x;
    int stride = gridDim.x * blockDim.x;
    float4 z = make_float4(0.f, 0.f, 0.f, 0.f);
    for (int j = i; j < nOut4; j += stride) out4[j] = z;
    for (int j = i; j < N; j += stride) { m_u[j] = 0u; s[j] = 0.0f; }
}

// ---------------- phase 1: edge scores via V_WMMA_F32_16X16X4_F32 ----------
// One wave32 handles 16 edges. A = 16x4 tile of concatenated [h_src0; h_dst0],
// B = w chunk broadcast across all 16 columns -> every column of the 16x16
// f32 accumulator holds that edge's raw attention score (exact f32).
__global__ void k_edge_scores(const float* __restrict__ h,
                              const int*  __restrict__ src,
                              const int*  __restrict__ dst,
                              const float* __restrict__ watt,
                              float* __restrict__ abuf,
                              unsigned* __restrict__ m_u,
                              int E, int ntiles) {
    __shared__ float wlds[2 * DIM];
    int tid = threadIdx.x;
    if (tid < 2 * DIM) wlds[tid] = watt[tid];
    __syncthreads();

    int tile = blockIdx.x * (blockDim.x >> 5) + (tid >> 5);
    if (tile >= ntiles) return;                 // uniform per wave

    int lane = tid & 31;
    int half = lane >> 4;                       // lane half: 0 or 1
    int m16  = lane & 15;                       // matrix row this lane feeds
    int base = tile * 16;

    int eL = base + m16; if (eL >= E) eL = E - 1;
    int sv = src[eL];
    int dv = dst[eL];
    // A slot rule (ISA 7.12.2, 32-bit 16x4 A): K = 2*half + vgpr
    const float* ps = h + (size_t)sv * ROW + 2 * half;   // h[src,0,:]
    const float* pd = h + (size_t)dv * ROW + 2 * half;   // h[dst,0,:]
    const float* wl = wlds + 2 * half;

    v8f c = {};
#if __has_builtin(__builtin_amdgcn_wmma_f32_16x16x4_f32)
    #pragma unroll
    for (int k = 0; k < 32; ++k) {              // K = 0..127 : src half
        v2f av = *(const v2f*)(ps + 4 * k);
        v2f bv = *(const v2f*)(wl + 4 * k);
        c = __builtin_amdgcn_wmma_f32_16x16x4_f32(false, av, false, bv,
                                                  (short)0, c, false, false);
    }
    #pragma unroll
    for (int k = 0; k < 32; ++k) {              // K = 128..255 : dst half
        v2f av = *(const v2f*)(pd + 4 * k);
        v2f bv = *(const v2f*)(wl + DIM + 4 * k);
        c = __builtin_amdgcn_wmma_f32_16x16x4_f32(false, av, false, bv,
                                                  (short)0, c, false, false);
    }
    // C layout: lane L holds rows {v + 8*(L>>4)} at column L&15; all columns
    // are equal (B was column-broadcast), so a select chain extracts row r.
    int r = lane & 7;
    float sc = c[0];
    sc = (r == 1) ? c[1] : sc;
    sc = (r == 2) ? c[2] : sc;
    sc = (r == 3) ? c[3] : sc;
    sc = (r == 4) ? c[4] : sc;
    sc = (r == 5) ? c[5] : sc;
    sc = (r == 6) ? c[6] : sc;
    sc = (r == 7) ? c[7] : sc;
#else
    // scalar fallback (kept only for compile portability)
    int r = lane & 7;
    int rowF = ((lane >> 4) << 3) | r;
    int eF = base + rowF; if (eF >= E) eF = E - 1;
    const float* fs = h + (size_t)src[eF] * ROW;
    const float* fd = h + (size_t)dst[eF] * ROW;
    float sc = 0.f;
    for (int k = 0; k < DIM; ++k) sc += fs[k] * wlds[k] + fd[k] * wlds[DIM + k];
#endif

    int row = ((lane >> 4) << 3) | r;           // lanes 0-7 -> rows 0-7, 16-23 -> 8-15
    if ((lane & 8) == 0) {
        int e = base + row;
        if (e < E) {
            float v = (sc >= 0.0f) ? sc : SLOPE * sc;   // leaky_relu
            abuf[e] = v;
            atomicMax(&m_u[dst[e]], flip_f32(v));
        }
    }
}

// ---------------- phase 2: w = exp(a - max), accumulate denominator --------
__global__ void k_edge_exp(const int* __restrict__ dst,
                           float* __restrict__ abuf,
                           const unsigned* __restrict__ m_u,
                           float* __restrict__ s, int E) {
    int e = blockIdx.x * blockDim.x + threadIdx.x;
    if (e >= E) return;
    int d = dst[e];
    float w = __expf(abuf[e] - unflip_f32(m_u[d]));
    abuf[e] = w;
    atomicAdd(&s[d], w);
}

// ---------------- phase 3: weighted scatter of full feature slabs ----------
__global__ void k_scatter(const float* __restrict__ h,
                          const int*  __restrict__ src,
                          const int*  __restrict__ dst,
                          const float* __restrict__ wbuf,
                          const float* __restrict__ s,
                          float* __restrict__ out, int E) {
    long gid = (long)blockIdx.x * blockDim.x + threadIdx.x;
    long tot = (long)E * (ROW / 4);
    if (gid >= tot) return;
    int e  = (int)(gid >> 6);                   // ROW/4 = 64 float4 per edge
    int c4 = (int)(gid & 63);
    int d  = dst[e];
    float alpha = wbuf[e] / s[d];
    const float4 v = *(const float4*)(h + (size_t)src[e] * ROW + c4 * 4);
    float* o = out + (size_t)d * ROW + c4 * 4;
    atomicAdd(o + 0, alpha * v.x);
    atomicAdd(o + 1, alpha * v.y);
    atomicAdd(o + 2, alpha * v.z);
    atomicAdd(o + 3, alpha * v.w);
}

// ---------------- phase 4: zero-degree nodes keep their old features -------
__global__ void k_finalize(const float* __restrict__ h,
                           const float* __restrict__ s,
                           float* __restrict__ out, int N) {
    long gid = (long)blockIdx.x * blockDim.x + threadIdx.x;
    long tot = (long)N * (ROW / 4);
    if (gid >= tot) return;
    int n  = (int)(gid >> 6);
    int c4 = (int)(gid & 63);
    if (s[n] == 0.0f) {     // no incoming edges (else s >= exp(0) = 1)
        *(float4*)(out + (size_t)n * ROW + c4 * 4) =
            *(const float4*)(h + (size_t)n * ROW + c4 * 4);
    }
}

extern "C" void kernel_launch(void* const* d_in, const int* in_sizes, int n_in,
                              void* d_out, int out_size, void* d_ws, size_t ws_size,
                              hipStream_t stream) {
    const float* h   = (const float*)d_in[0];  // [N, 2, 128]
    const int*   src = (const int*)d_in[1];    // [E]
    const int*   dst = (const int*)d_in[2];    // [E]
    const float* w   = (const float*)d_in[3];  // [1, 256]
    int E = in_sizes[1];
    int N = in_sizes[0] / ROW;
    float* out = (float*)d_out;

    // workspace: a/w per edge, per-node max (flipped u32) and sum  (~3.6 MB)
    float*    abuf = (float*)d_ws;
    unsigned* m_u  = (unsigned*)((char*)d_ws + (size_t)E * 4);
    float*    s    = (float*)   ((char*)d_ws + (size_t)E * 4 + (size_t)N * 4);

    int nOut4 = N * (ROW / 4);
    k_init<<<1024, 256, 0, stream>>>((float4*)out, nOut4, m_u, s, N);

    int ntiles = (E + 15) / 16;
    k_edge_scores<<<(ntiles + 7) / 8, 256, 0, stream>>>(h, src, dst, w,
                                                        abuf, m_u, E, ntiles);

    k_edge_exp<<<(E + 255) / 256, 256, 0, stream>>>(dst, abuf, m_u, s, E);

    long tot4 = (long)E * (ROW / 4);
    k_scatter<<<(int)((tot4 + 255) / 256), 256, 0, stream>>>(h, src, dst,
                                                             abuf, s, out, E);

    long totf = (long)N * (ROW / 4);
    k_finalize<<<(int)((totf + 255) / 256), 256, 0, stream>>>(h, s, out, N);
}